// CRF_21784074125389
// MI455X (gfx1250) — compile-verified
//
#include <hip/hip_runtime.h>
#include <math.h>

#define Bn 512
#define Tn 2048
#define Kn 26
#define NEG_INF (-__builtin_inff())
#define LOG2E 1.4426950408889634f
#define LN2   0.6931471805599453f

typedef float v2f __attribute__((ext_vector_type(2)));
typedef float v8f __attribute__((ext_vector_type(8)));

// LDS: viterbi branch needs 26*2052 (backptr bytes) + 2048 (mask) + 2048 (path)
// fwd branch aliases the front of it for expT (28*32 f32) + pbuf (32*17 f32)
#define SMEM_BYTES (26 * 2052 + 2048 + 2048)

__global__ __launch_bounds__(32) void crf_main(
    const float* __restrict__ fc, const int* __restrict__ tags,
    const float* __restrict__ startT, const float* __restrict__ endT,
    const float* __restrict__ trans, float* __restrict__ out,
    float* __restrict__ logZ, float* __restrict__ scoreN)
{
    __shared__ __align__(16) unsigned char smem[SMEM_BYTES];

    if (blockIdx.x < Bn) {
        // ================= Viterbi + gold-path score: one wave per batch ====
        unsigned char* hist = smem;                  // [26][2052]
        unsigned char* msk  = smem + 26 * 2052;      // [2048]
        unsigned char* pth  = msk + 2048;            // [2048]
        const int b = blockIdx.x;
        const int k = threadIdx.x;
        const bool act = k < Kn;
        const int kc = act ? k : 0;                  // clamped, branchless loads

        float tcol[Kn];
#pragma unroll
        for (int kp = 0; kp < Kn; ++kp)
            tcol[kp] = trans[kp * Kn + kc];

        float score = act ? (startT[kc] + fc[(size_t)b * Tn * Kn + kc]) : NEG_INF;

        // software-pipelined scan: prefetch t+1 while computing t
        int   tg  = tags[b * Tn + 1];
        float etr = fc[((size_t)b * Tn + 1) * Kn + kc];
#pragma unroll 2
        for (int t = 1; t < Tn; ++t) {
            const int tn = (t + 1 < Tn) ? (t + 1) : (Tn - 1);
            const int   tg_n  = tags[b * Tn + tn];
            const float etr_n = fc[((size_t)b * Tn + tn) * Kn + kc];

            const bool  mt = (tg != 0);
            const float et = act ? etr : NEG_INF;
            float best = NEG_INF;
            int arg = 0;
#pragma unroll
            for (int kp = 0; kp < Kn; ++kp) {
                float s = __shfl(score, kp, 32);
                float cand = s + tcol[kp];
                if (cand > best) { best = cand; arg = kp; }  // first-max ties
            }
            if (act) hist[k * 2052 + t] = (unsigned char)arg;
            if (k == 0) msk[t] = mt ? 1 : 0;
            score = mt ? (best + et) : score;
            tg = tg_n; etr = etr_n;
        }
        __syncthreads();

        // argmax(score + end) across lanes, lowest index on ties
        float bv = score + (act ? endT[kc] : NEG_INF);
        int bi = k;
#pragma unroll
        for (int s = 1; s < 32; s <<= 1) {
            float ov = __shfl_xor(bv, s, 32);
            int oi = __shfl_xor(bi, s, 32);
            if (ov > bv || (ov == bv && oi < bi)) { bv = ov; bi = oi; }
        }
        if (k == 0) {
            int cur = bi;
            for (int t = Tn - 1; t >= 1; --t) {
                bool mt = msk[t] != 0;
                pth[t] = mt ? (unsigned char)cur : (unsigned char)0;  // PAD=0
                if (mt) cur = hist[cur * 2052 + t];
            }
            pth[0] = (unsigned char)cur;
        }
        __syncthreads();
        for (int t = k; t < Tn; t += 32)
            out[b * Tn + t] = (float)pth[t];

        // ---- gold path score (parallel over t, shfl reduce) ----
        float part = 0.0f;
        int lastpos = 0;
        for (int t = k; t < Tn; t += 32) {
            int tgv = tags[b * Tn + t];
            if (t == 0) {
                part += startT[tgv] + fc[(size_t)b * Tn * Kn + tgv];
            } else if (tgv != 0) {
                int tp = tags[b * Tn + t - 1];
                part += trans[tp * Kn + tgv] + fc[((size_t)b * Tn + t) * Kn + tgv];
                lastpos = (t > lastpos) ? t : lastpos;
            }
        }
#pragma unroll
        for (int s = 1; s < 32; s <<= 1) {
            part += __shfl_xor(part, s, 32);
            int ol = __shfl_xor(lastpos, s, 32);
            lastpos = (ol > lastpos) ? ol : lastpos;
        }
        if (k == 0)
            scoreN[b] = part + endT[tags[b * Tn + lastpos]];

    } else {
        // ======= log-partition forward via fp32 WMMA: 16 batch rows / wave ==
        float* expT = (float*)smem;           // [28][32] exp(trans), 0-padded
        float* pbuf = expT + 28 * 32;         // p column-major, stride 17
        const int lane = threadIdx.x;
        const int b0 = (blockIdx.x - Bn) * 16;
        const int hi = lane >> 4;
        const int lo = lane & 15;
        const int col0 = lo;                  // tag column, tile 0
        const int col1 = 16 + lo;             // tag column, tile 1
        const bool v1 = (col1 < Kn);
        const int c1 = v1 ? col1 : 0;         // clamped, branchless loads

        for (int i = lane; i < 28 * 32; i += 32) {
            int kp = i >> 5, kk = i & 31;
            expT[i] = (kp < Kn && kk < Kn) ? __expf(trans[kp * Kn + kk]) : 0.0f;
        }
        __syncthreads();

        // B operand in registers: VGPR v, lane L <-> (k' = 4c + 2*hi + v, n)
        v2f Bt[2][7];
#pragma unroll
        for (int n = 0; n < 2; ++n)
#pragma unroll
            for (int c = 0; c < 7; ++c) {
                int kp = 4 * c + 2 * hi;
                int kk = n * 16 + lo;
                Bt[n][c].x = expT[kp * 32 + kk];
                Bt[n][c].y = expT[(kp + 1) * 32 + kk];
            }

        // alpha in C/D layout: al[n][j] = alpha[row = b0+j+8*hi, col = n*16+lo]
        float al[2][8];
        const float sc0 = startT[col0];
        const float sc1 = startT[c1];
#pragma unroll
        for (int j = 0; j < 8; ++j) {
            int row = b0 + j + 8 * hi;
            size_t base = (size_t)row * Tn * Kn;
            al[0][j] = sc0 + fc[base + col0];
            float r1 = sc1 + fc[base + c1];
            al[1][j] = v1 ? r1 : NEG_INF;
        }

        // emission/tag prefetch registers (loads issue early, ballot at use)
        float ce0[8], ce1r[8];
        int ctag;
        {
            ctag = tags[(b0 + lo) * Tn + 1];
#pragma unroll
            for (int j = 0; j < 8; ++j) {
                size_t base = ((size_t)(b0 + j + 8 * hi) * Tn + 1) * Kn;
                ce0[j]  = fc[base + col0];
                ce1r[j] = fc[base + c1];
            }
        }

#pragma unroll 2
        for (int t = 1; t < Tn; ++t) {
            // prefetch step t+1 (clamped at the end; redundant last iter)
            const int tn = (t + 1 < Tn) ? (t + 1) : (Tn - 1);
            float ne0[8], ne1r[8];
            int ntag = tags[(b0 + lo) * Tn + tn];
#pragma unroll
            for (int j = 0; j < 8; ++j) {
                size_t base = ((size_t)(b0 + j + 8 * hi) * Tn + tn) * Kn;
                ne0[j]  = fc[base + col0];
                ne1r[j] = fc[base + c1];
            }

            // per-row max (butterfly in 16-lane halves); p = 2^((al-mx)*log2e)
            // via raw v_exp_f32 (inputs <= 0, never denormal-problematic)
            float mxv[8];
#pragma unroll
            for (int j = 0; j < 8; ++j) {
                float m = fmaxf(al[0][j], al[1][j]);
#pragma unroll
                for (int s = 1; s < 16; s <<= 1)
                    m = fmaxf(m, __shfl_xor(m, s, 32));
                mxv[j] = m;
                pbuf[col0 * 17 + j + 8 * hi] =
                    __builtin_amdgcn_exp2f((al[0][j] - m) * LOG2E);
                pbuf[col1 * 17 + j + 8 * hi] =
                    __builtin_amdgcn_exp2f((al[1][j] - m) * LOG2E);
            }
            __syncthreads();

            // A-layout loads + 14x V_WMMA_F32_16X16X4_F32
            v8f acc0 = {};
            v8f acc1 = {};
#pragma unroll
            for (int c = 0; c < 7; ++c) {
                int kp = 4 * c + 2 * hi;
                v2f A;
                A.x = pbuf[kp * 17 + lo];
                A.y = pbuf[(kp + 1) * 17 + lo];
                acc0 = __builtin_amdgcn_wmma_f32_16x16x4_f32(
                    false, A, false, Bt[0][c], (short)0, acc0, false, false);
                acc1 = __builtin_amdgcn_wmma_f32_16x16x4_f32(
                    false, A, false, Bt[1][c], (short)0, acc1, false, false);
            }
            __syncthreads();

            // al' = log2(acc)*ln2 + mx + e  (raw v_log_f32; acc >= ~0.9 or 0)
            const unsigned mb = (unsigned)__ballot(ctag != 0);
#pragma unroll
            for (int j = 0; j < 8; ++j) {
                bool mt = (mb >> (j + 8 * hi)) & 1u;
                float e1 = v1 ? ce1r[j] : NEG_INF;
                float n0 = __builtin_amdgcn_logf(acc0[j]) * LN2 + mxv[j] + ce0[j];
                float n1 = __builtin_amdgcn_logf(acc1[j]) * LN2 + mxv[j] + e1;
                al[0][j] = mt ? n0 : al[0][j];
                al[1][j] = mt ? n1 : al[1][j];
                ce0[j] = ne0[j]; ce1r[j] = ne1r[j];
            }
            ctag = ntag;
        }

        // logZ[row] = logsumexp_k(alpha + end)
        const float ev0 = endT[col0];
        const float ev1 = v1 ? endT[c1] : NEG_INF;
        float res[8];
#pragma unroll
        for (int j = 0; j < 8; ++j) {
            float a0 = al[0][j] + ev0;
            float a1 = al[1][j] + ev1;
            float m = fmaxf(a0, a1);
#pragma unroll
            for (int s = 1; s < 16; s <<= 1)
                m = fmaxf(m, __shfl_xor(m, s, 32));
            float sum = __builtin_amdgcn_exp2f((a0 - m) * LOG2E) +
                        __builtin_amdgcn_exp2f((a1 - m) * LOG2E);
#pragma unroll
            for (int s = 1; s < 16; s <<= 1)
                sum += __shfl_xor(sum, s, 32);
            res[j] = m + __builtin_amdgcn_logf(sum) * LN2;
        }
        if (lane == 0) {
#pragma unroll
            for (int j = 0; j < 8; ++j) logZ[b0 + j] = res[j];
        }
        if (lane == 16) {
#pragma unroll
            for (int j = 0; j < 8; ++j) logZ[b0 + 8 + j] = res[j];
        }
    }
}

__global__ __launch_bounds__(256) void crf_loss(
    const float* __restrict__ logZ, const float* __restrict__ scoreN,
    float* __restrict__ out)
{
    __shared__ float sbuf[256];
    int tid = threadIdx.x;
    float s = 0.0f;
    for (int b = tid; b < Bn; b += 256) s += logZ[b] - scoreN[b];
    sbuf[tid] = s;
    __syncthreads();
    for (int st = 128; st > 0; st >>= 1) {
        if (tid < st) sbuf[tid] += sbuf[tid + st];
        __syncthreads();
    }
    if (tid == 0) out[Bn * Tn] = sbuf[0] / (float)Bn;
}

extern "C" void kernel_launch(void* const* d_in, const int* in_sizes, int n_in,
                              void* d_out, int out_size, void* d_ws, size_t ws_size,
                              hipStream_t stream) {
    const float* fc     = (const float*)d_in[0];
    const int*   tags   = (const int*)d_in[1];
    const float* startT = (const float*)d_in[2];
    const float* endT   = (const float*)d_in[3];
    const float* trans  = (const float*)d_in[4];
    float* out    = (float*)d_out;
    float* logZ   = (float*)d_ws;        // [512]
    float* scoreN = logZ + Bn;           // [512]

    // 512 viterbi/numerator blocks + 32 WMMA-forward blocks in one launch
    crf_main<<<Bn + Bn / 16, 32, 0, stream>>>(fc, tags, startT, endT, trans,
                                              out, logZ, scoreN);
    crf_loss<<<1, 256, 0, stream>>>(logZ, scoreN, out);
}